// Filter_high_35828617183256
// MI455X (gfx1250) — compile-verified
//
#include <hip/hip_runtime.h>
#include <hip/hip_bf16.h>
#include <stdint.h>

typedef float v2f __attribute__((ext_vector_type(2)));
typedef float v8f __attribute__((ext_vector_type(8)));

#define BB 32
#define CC 256
#define NN 8192

// ---------------------------------------------------------------------------
// Kernel A: squared L2 norms of 16-column groups via WMMA Gram diagonal.
// A (16x4 f32, lane=M, vgpr=K) and B (4x16 f32, lane=N, vgpr=K) register
// images of the same tile are identical, so one v2f load pair serves both.
// Writes sort keys: (bits(norm2)<<32) | ~column  (stable-descending order).
// ---------------------------------------------------------------------------
__global__ __launch_bounds__(256) void norm_kernel(const float* __restrict__ x,
                                                   unsigned long long* __restrict__ keys) {
    const int lane = threadIdx.x & 31;
    const int wave = threadIdx.x >> 5;
    const int tile = blockIdx.x * 8 + wave;     // 32*512 = 16384 tiles
    const int b    = tile >> 9;                 // tile / 512
    const int n0   = (tile & 511) << 4;         // 16 columns per tile

    const int half = lane >> 4;                 // 0 -> K=0,1 ; 1 -> K=2,3
    const int m    = lane & 15;                 // column (M / N) within tile

    const float* p = x + (size_t)b * CC * NN + (size_t)(2 * half) * NN + (n0 + m);

    v8f acc = {};
    #pragma unroll 4
    for (int k = 0; k < 64; ++k) {              // 64 chunks * K=4 -> K=256
        v2f a;
        a.x = p[0];
        a.y = p[NN];
        p += (size_t)4 * NN;
        // D += A * B with A == B  => Gram matrix of 16 columns
        acc = __builtin_amdgcn_wmma_f32_16x16x4_f32(false, a, false, a,
                                                    (short)0, acc, false, false);
    }

    // Diagonal D[m][m]: m<8 -> vgpr m, lane m ; m>=8 -> vgpr m-8, lane m+16.
    int sel = lane & 7;
    float a01 = (sel & 1) ? acc[1] : acc[0];
    float a23 = (sel & 1) ? acc[3] : acc[2];
    float a45 = (sel & 1) ? acc[5] : acc[4];
    float a67 = (sel & 1) ? acc[7] : acc[6];
    float b03 = (sel & 2) ? a23 : a01;
    float b47 = (sel & 2) ? a67 : a45;
    float diag = (sel & 4) ? b47 : b03;

    bool active = (lane < 8) || (lane >= 24);
    int  mcol   = (lane < 8) ? lane : (lane - 16);
    if (active) {
        unsigned n = (unsigned)(n0 + mcol);
        unsigned long long key =
            ((unsigned long long)__float_as_uint(diag) << 32) |
            (unsigned long long)(0xFFFFFFFFu - n);
        keys[(size_t)b * NN + n] = key;
    }
}

// ---------------------------------------------------------------------------
// Kernel B: per-batch descending bitonic sort of 8192 packed keys in LDS,
// then emit the top-capsnum original column indices.
// ---------------------------------------------------------------------------
__global__ __launch_bounds__(1024) void sort_kernel(const unsigned long long* __restrict__ keys,
                                                    int* __restrict__ order, int capsnum) {
    __shared__ unsigned long long s[NN];        // 64 KB of the 320 KB WGP LDS
    const int b   = blockIdx.x;
    const int tid = threadIdx.x;

    for (int i = tid; i < NN; i += 1024) s[i] = keys[(size_t)b * NN + i];
    __syncthreads();

    for (int k = 2; k <= NN; k <<= 1) {
        for (int j = k >> 1; j > 0; j >>= 1) {
            for (int i = tid; i < NN; i += 1024) {
                int ixj = i ^ j;
                if (ixj > i) {
                    unsigned long long va = s[i], vb = s[ixj];
                    bool up = (i & k) == 0;
                    bool sw = up ? (va < vb) : (va > vb);   // flipped => descending
                    if (sw) { s[i] = vb; s[ixj] = va; }
                }
            }
            __syncthreads();
        }
    }

    for (int i = tid; i < capsnum; i += 1024)
        order[(size_t)b * capsnum + i] =
            (int)(0xFFFFFFFFu - (unsigned)(s[i] & 0xFFFFFFFFull));
}

// ---------------------------------------------------------------------------
// Kernel C: out[b,c,j] = x[b,c,order[b,j]].  Coalesced writes; scattered
// reads stay inside one 32 KB row of x per (b,c).
// ---------------------------------------------------------------------------
__global__ __launch_bounds__(256) void gather_kernel(const float* __restrict__ x,
                                                     const int* __restrict__ order,
                                                     float* __restrict__ out, int capsnum) {
    const int j  = blockIdx.x * 256 + threadIdx.x;
    const int bc = blockIdx.y;                  // b*256 + c
    if (j >= capsnum) return;
    const int b = bc >> 8;
    __builtin_prefetch(order + (size_t)b * capsnum + j + 4096, 0, 1);
    int n = order[(size_t)b * capsnum + j];
    out[(size_t)bc * capsnum + j] = x[(size_t)bc * NN + n];
}

// ---------------------------------------------------------------------------
extern "C" void kernel_launch(void* const* d_in, const int* in_sizes, int n_in,
                              void* d_out, int out_size, void* d_ws, size_t ws_size,
                              hipStream_t stream) {
    const float* x  = (const float*)d_in[0];
    float* out      = (float*)d_out;
    const int capsnum = out_size / (BB * CC);   // reference's data-dependent cut, fixed by harness

    // keys (2 MB) staged in d_out: dead before gather fully overwrites d_out.
    unsigned long long* keys = (unsigned long long*)d_out;
    int* order = (int*)d_ws;                    // 32*capsnum ints (< 1 MB)

    norm_kernel<<<2048, 256, 0, stream>>>(x, keys);
    sort_kernel<<<BB, 1024, 0, stream>>>(keys, order, capsnum);
    dim3 g((capsnum + 255) / 256, BB * CC);
    gather_kernel<<<g, 256, 0, stream>>>(x, order, out, capsnum);
}